// QModel_65481071404032
// MI455X (gfx1250) — compile-verified
//
#include <hip/hip_runtime.h>

#define NW 18
#define DIM (1 << 18)
#define BATCH 128
#define TILE 32768      // 2^15 amplitudes per LDS tile (wires 3..17 local)
#define CHUNK_B 64      // batch chunk that fits in 192MB L2 (64 * 2MB = 128MB)

typedef __attribute__((ext_vector_type(2))) float v2f;
typedef __attribute__((ext_vector_type(8))) float v8f;

struct c2 { float re, im; };
__device__ __forceinline__ c2 cmul(c2 a, c2 b) {
    return c2{a.re * b.re - a.im * b.im, a.re * b.im + a.im * b.re};
}

// U = RZ(z) @ RY(y) @ RX(x), row-major 2x2 complex
__device__ __forceinline__ void gate_u(const float* p, c2 U[2][2]) {
    float cx = cosf(0.5f * p[0]), sx = sinf(0.5f * p[0]);
    float cy = cosf(0.5f * p[1]), sy = sinf(0.5f * p[1]);
    float cz = cosf(0.5f * p[2]), sz = sinf(0.5f * p[2]);
    c2 m00{ cy * cx,  sy * sx};
    c2 m01{-sy * cx, -cy * sx};
    c2 m10{ sy * cx, -cy * sx};
    c2 m11{ cy * cx, -sy * sx};
    c2 ez{cz, -sz}, ezc{cz, sz};   // e^{-iz/2}, e^{+iz/2}
    U[0][0] = cmul(ez, m00);  U[0][1] = cmul(ez, m01);
    U[1][0] = cmul(ezc, m10); U[1][1] = cmul(ezc, m11);
}

// ---------------------------------------------------------------------------
// Setup: per-layer 2x2 gate tables (wires 3..17) and the real 16x16 matrix M
// for wires 0..2 (8x8 complex kron -> real form on [re;im]-interleaved rows).
// ---------------------------------------------------------------------------
__global__ void k_setup(const float* __restrict__ params,
                        float* __restrict__ gates,   // [3][18][8]
                        float* __restrict__ M16) {   // [3][16][16]
    int tid = threadIdx.x;
    if (tid < 3 * NW) {
        int l = tid / NW, w = tid % NW;
        c2 U[2][2];
        gate_u(params + (l * NW + w) * 3, U);
        float* g = gates + (l * NW + w) * 8;
        g[0] = U[0][0].re; g[1] = U[0][0].im; g[2] = U[0][1].re; g[3] = U[0][1].im;
        g[4] = U[1][0].re; g[5] = U[1][0].im; g[6] = U[1][1].re; g[7] = U[1][1].im;
    }
    for (int idx = tid; idx < 3 * 256; idx += blockDim.x) {
        int l = idx >> 8, m = (idx >> 4) & 15, k = idx & 15;
        c2 U0[2][2], U1[2][2], U2[2][2];
        gate_u(params + (l * NW + 0) * 3, U0);
        gate_u(params + (l * NW + 1) * 3, U1);
        gate_u(params + (l * NW + 2) * 3, U2);
        int j = m >> 1, ci = m & 1, jp = k >> 1, cj = k & 1;
        // j bit2<->wire0 (state bit17), bit1<->wire1, bit0<->wire2
        c2 W = cmul(cmul(U0[(j >> 2) & 1][(jp >> 2) & 1],
                         U1[(j >> 1) & 1][(jp >> 1) & 1]),
                    U2[j & 1][jp & 1]);
        float val = (ci == 0) ? ((cj == 0) ? W.re : -W.im)
                              : ((cj == 0) ? W.im :  W.re);
        M16[l * 256 + m * 16 + k] = val;
    }
}

// ---------------------------------------------------------------------------
// Pass 1: 15 local gates (wires 3..17) on a 2^15 tile. 32 amps/thread in
// registers, two LDS re-stagings. XOR swizzle keeps LDS conflict-free.
// ---------------------------------------------------------------------------
__device__ __forceinline__ int sw(int x) { return x ^ ((x >> 5) & 31); }

template <int WB>
__device__ __forceinline__ void apply5(float2 v[32], const float* gl) {
#pragma unroll
    for (int k = 0; k < 5; ++k) {            // gate on wire (WB - k), e-bit k
        const float* g = gl + (WB - k) * 8;
        float ar = g[0], ai = g[1], br = g[2], bi = g[3];
        float cr = g[4], ci = g[5], dr = g[6], di = g[7];
#pragma unroll
        for (int e0 = 0; e0 < 32; ++e0) {
            if (e0 & (1 << k)) continue;
            int e1 = e0 | (1 << k);
            float2 x0 = v[e0], x1 = v[e1];
            v[e0] = make_float2(ar * x0.x - ai * x0.y + br * x1.x - bi * x1.y,
                                ar * x0.y + ai * x0.x + br * x1.y + bi * x1.x);
            v[e1] = make_float2(cr * x0.x - ci * x0.y + dr * x1.x - di * x1.y,
                                cr * x0.y + ci * x0.x + dr * x1.y + di * x1.x);
        }
    }
}

__global__ void __launch_bounds__(1024) k_local(
    const float* __restrict__ srcR, const float* __restrict__ srcI,
    float2* state, const float* __restrict__ gl, int b0, int planar) {
    extern __shared__ float2 lds[];          // TILE float2 = 256KB
    int t = threadIdx.x;
    int bb = b0 + (blockIdx.x >> 3);
    int j  = blockIdx.x & 7;
    size_t gbase = ((size_t)bb << 18) + ((size_t)j << 15);

    float2 v[32];
    // Stage A: x = e<<10 | t  (e-bit k <-> state bit 10+k <-> wire 7-k)
#pragma unroll
    for (int e = 0; e < 32; ++e) {
        int x = (e << 10) | t;
        if (planar) v[e] = make_float2(srcR[gbase + x], srcI[gbase + x]);
        else        v[e] = state[gbase + x];
    }
    apply5<7>(v, gl);                        // wires 7..3
#pragma unroll
    for (int e = 0; e < 32; ++e) lds[sw((e << 10) | t)] = v[e];
    __syncthreads();
    // Stage B: x = t<<5 | e  (e-bit k <-> bit k <-> wire 17-k)
#pragma unroll
    for (int e = 0; e < 32; ++e) v[e] = lds[sw((t << 5) | e)];
    apply5<17>(v, gl);                       // wires 17..13
#pragma unroll
    for (int e = 0; e < 32; ++e) lds[sw((t << 5) | e)] = v[e];
    __syncthreads();
    // Stage C: x = (t>>5)<<10 | e<<5 | (t&31)  (e-bit k <-> bit 5+k <-> wire 12-k)
#pragma unroll
    for (int e = 0; e < 32; ++e)
        v[e] = lds[sw(((t >> 5) << 10) | (e << 5) | (t & 31))];
    apply5<12>(v, gl);                       // wires 12..8
#pragma unroll
    for (int e = 0; e < 32; ++e) {
        int x = ((t >> 5) << 10) | (e << 5) | (t & 31);
        state[gbase + x] = v[e];             // lanes consecutive -> coalesced
    }
}

// ---------------------------------------------------------------------------
// Pass 2: wires 0..2 as Y = M(16x16) X via v_wmma_f32_16x16x4_f32 (4 chained
// K-chunks). Non-final: apply CZ sign diagonal + store. Final: fused readout.
// ---------------------------------------------------------------------------
template <int FINAL>
__global__ void __launch_bounds__(256) k_global(
    float2* state, const float* __restrict__ Mp,
    const float* __restrict__ head_w, float* __restrict__ partials, int b0) {
    int l    = threadIdx.x & 31;
    int wv   = threadIdx.x >> 5;
    int blk  = blockIdx.x & 31;
    int bb   = b0 + (blockIdx.x >> 5);
    size_t gbase = (size_t)bb << 18;
    int half = l >> 4;                 // lane half selects K offset / row half
    int koff = half * 2;
    int m    = l & 15;

    v2f a[4];                          // A fragments: M rows, K-pairs per half
#pragma unroll
    for (int k = 0; k < 4; ++k) {
        a[k].x = Mp[m * 16 + 4 * k + koff];
        a[k].y = Mp[m * 16 + 4 * k + koff + 1];
    }
    float hw[NW];
    if (FINAL) {
#pragma unroll
        for (int i = 0; i < NW; ++i) hw[i] = head_w[i];
    }
    float accS = 0.0f;

    for (int it = 0; it < 8; ++it) {
        int col = ((blk * 64 + it * 8 + wv) << 4) + (l & 15);
        v8f acc = {0.f, 0.f, 0.f, 0.f, 0.f, 0.f, 0.f, 0.f};
#pragma unroll
        for (int k = 0; k < 4; ++k) {
            int jj = 2 * k + half;     // amplitude row owned by this lane/chunk
            float2 xv = state[gbase + ((size_t)jj << 15) + col];
            v2f bf; bf.x = xv.x; bf.y = xv.y;   // rows 2jj (re) / 2jj+1 (im)
            acc = __builtin_amdgcn_wmma_f32_16x16x4_f32(
                false, a[k], false, bf, (short)0, acc, false, false);
        }
        if (FINAL) {
            float clo = 0.0f;          // head_w contribution of low 15 bits
#pragma unroll
            for (int b2 = 0; b2 < 15; ++b2) {
                float h = hw[17 - b2];
                clo += ((col >> b2) & 1) ? -h : h;
            }
#pragma unroll
            for (int jp = 0; jp < 4; ++jp) {
                int amp = half * 4 + jp;
                float re = acc[2 * jp], im = acc[2 * jp + 1];
                float chi = ((amp & 4) ? -hw[0] : hw[0])
                          + ((amp & 2) ? -hw[1] : hw[1])
                          + ((amp & 1) ? -hw[2] : hw[2]);
                accS += (re * re + im * im) * (clo + chi);
            }
        } else {
#pragma unroll
            for (int jp = 0; jp < 4; ++jp) {
                int amp = half * 4 + jp;
                int idx = (amp << 15) | col;
                int sgn = __popc(idx & (idx >> 1) & 0x1FFFF) & 1;  // all CZs
                float re = acc[2 * jp], im = acc[2 * jp + 1];
                state[gbase + idx] = sgn ? make_float2(-re, -im)
                                         : make_float2(re, im);
            }
        }
    }
    if (FINAL) {
        __shared__ float red[256];
        int tid = threadIdx.x;
        red[tid] = accS;
        __syncthreads();
#pragma unroll
        for (int s = 128; s > 0; s >>= 1) {
            if (tid < s) red[tid] += red[tid + s];
            __syncthreads();
        }
        if (tid == 0) partials[bb * 32 + blk] = red[0];
    }
}

__global__ void k_reduce(const float* __restrict__ partials,
                         const float* __restrict__ head_b,
                         float* __restrict__ out) {
    int b = threadIdx.x;               // 128 threads
    float s = 0.0f;
#pragma unroll
    for (int k = 0; k < 32; ++k) s += partials[b * 32 + k];
    out[b] = s + head_b[0];
}

// ---------------------------------------------------------------------------
extern "C" void kernel_launch(void* const* d_in, const int* in_sizes, int n_in,
                              void* d_out, int out_size, void* d_ws, size_t ws_size,
                              hipStream_t stream) {
    const float* sr     = (const float*)d_in[0];
    const float* si     = (const float*)d_in[1];
    const float* params = (const float*)d_in[2];
    const float* head_w = (const float*)d_in[3];
    const float* head_b = (const float*)d_in[4];
    float* out = (float*)d_out;

    const size_t STATE_BYTES = (size_t)BATCH * DIM * sizeof(float2); // 256MB
    char* ws = (char*)d_ws;
    float2* state   = (float2*)ws;
    float* gates    = (float*)(ws + STATE_BYTES);           // 3*18*8 floats
    float* M16      = (float*)(ws + STATE_BYTES + 4096);    // 3*256 floats
    float* partials = (float*)(ws + STATE_BYTES + 8192);    // 128*32 floats

    k_setup<<<1, 256, 0, stream>>>(params, gates, M16);

    for (int c = 0; c < BATCH / CHUNK_B; ++c) {
        int b0 = c * CHUNK_B;
        for (int layer = 0; layer < 3; ++layer) {
            k_local<<<CHUNK_B * 8, 1024, TILE * sizeof(float2), stream>>>(
                sr, si, state, gates + layer * NW * 8, b0, layer == 0 ? 1 : 0);
            if (layer < 2)
                k_global<0><<<CHUNK_B * 32, 256, 0, stream>>>(
                    state, M16 + layer * 256, head_w, partials, b0);
            else
                k_global<1><<<CHUNK_B * 32, 256, 0, stream>>>(
                    state, M16 + layer * 256, head_w, partials, b0);
        }
    }
    k_reduce<<<1, 128, 0, stream>>>(partials, head_b, out);
}